// DFineDecoder_14413910245499
// MI455X (gfx1250) — compile-verified
//
#include <hip/hip_runtime.h>
#include <hip/hip_bf16.h>
#include <math.h>

// ---------------------------------------------------------------------------
// D-Fine decoder layer on MI455X (gfx1250, wave32).
// All dense GEMMs + attention matmuls use V_WMMA_F32_16X16X4_F32 (exact fp32
// matrix path). Irregular ops (softmax, LN, sigmoid gate, bilinear gather)
// are VALU wave32 kernels. Workspace arena ~112 MB.
// ---------------------------------------------------------------------------

typedef float v2f __attribute__((ext_vector_type(2)));
typedef float v8f __attribute__((ext_vector_type(8)));

#define Bc 32
#define Qc 300
#define Dc 256
#define NHc 8
#define HDc 32
#define LTOT 8400

__device__ inline v8f wmma_f32_4(v2f a, v2f b, v8f c) {
  return __builtin_amdgcn_wmma_f32_16x16x4_f32(
      /*neg_a=*/false, a, /*neg_b=*/false, b,
      /*c_mod=*/(short)0, c, /*reuse_a=*/false, /*reuse_b=*/false);
}

// ---------------------------------------------------------------------------
// Generic fp32 WMMA GEMM: C[M,N] = act(A[M,K] @ W[K,N] + bias) (+= if accum)
// Block: 256 threads = 8 waves; wave w owns C tile rows (by*8+w)*16, cols bx*16.
// B panel (16 cols) staged through LDS in K-chunks of 64.
// Requires: M % 128 == 0, N % 16 == 0, K % 64 == 0.
// ---------------------------------------------------------------------------
__global__ __launch_bounds__(256) void gemm_wmma_kernel(
    const float* __restrict__ A, int lda,
    const float* __restrict__ W, int ldw,
    const float* __restrict__ bias,
    float* __restrict__ C, int ldc,
    int M, int N, int K, int relu, int accum) {
  __shared__ float Bt[16][66];  // [n][k] transposed, padded stride 66

  const int lane  = threadIdx.x & 31;
  const int wave  = threadIdx.x >> 5;
  const int n16   = lane & 15;              // N index (B/C frag) and M index (A frag)
  const int khalf = (lane >> 4) << 1;       // 0 or 2 (K sub-offset per ISA layout)
  const int mhi   = (lane >> 4) << 3;       // +8 row offset for C frag upper half

  const int n0   = blockIdx.x * 16;
  const int row0 = (blockIdx.y * 8 + wave) * 16;
  if (row0 >= M) return;

  v8f c = {};
  if (accum) {
#pragma unroll
    for (int r = 0; r < 8; ++r)
      c[r] = C[(size_t)(row0 + r + mhi) * ldc + n0 + n16];
  }

  for (int kc = 0; kc < K; kc += 64) {
    // cooperative load of W[kc:kc+64, n0:n0+16] -> Bt[n][k]
#pragma unroll
    for (int i = threadIdx.x; i < 1024; i += 256) {
      int r = i >> 4, cc = i & 15;
      Bt[cc][r] = W[(size_t)(kc + r) * ldw + n0 + cc];
    }
    __syncthreads();

    const float* Arow = A + (size_t)(row0 + n16) * lda + kc + khalf;
    if (kc + 64 < K) __builtin_prefetch(Arow + 64, 0, 3);
    const float2* Ap = reinterpret_cast<const float2*>(Arow);
#pragma unroll
    for (int s = 0; s < 16; ++s) {
      float2 av = Ap[s * 2];
      v2f a; a.x = av.x; a.y = av.y;
      v2f b; b.x = Bt[n16][s * 4 + khalf]; b.y = Bt[n16][s * 4 + khalf + 1];
      c = wmma_f32_4(a, b, c);
    }
    __syncthreads();
  }

  float bv = bias ? bias[n0 + n16] : 0.0f;
#pragma unroll
  for (int r = 0; r < 8; ++r) {
    float val = c[r] + bv;
    if (relu) val = fmaxf(val, 0.0f);
    C[(size_t)(row0 + r + mhi) * ldc + n0 + n16] = val;
  }
}

// ---------------------------------------------------------------------------
// Self-attention: one wave per (b, h, 16-query tile). Scores via WMMA into an
// LDS scoreboard, softmax in LDS, ctx = P @ V via WMMA with A frags from LDS.
// q/k/v/ctx layout: (B*Q, 256), head h occupies cols h*32..h*32+31.
// ---------------------------------------------------------------------------
__global__ __launch_bounds__(32) void attn_kernel(
    const float* __restrict__ q, const float* __restrict__ k,
    const float* __restrict__ v, float* __restrict__ ctx) {
  __shared__ float S[16][312];  // 19 key tiles * 16 = 304 cols, padded

  const int tile = blockIdx.x;          // 19 * NH * B tiles
  const int mt = tile % 19;
  const int h  = (tile / 19) % NHc;
  const int b  = tile / (19 * NHc);
  const int q0 = mt * 16;

  const int lane  = threadIdx.x;
  const int n16   = lane & 15;
  const int khalf = (lane >> 4) << 1;
  const int mhi   = (lane >> 4) << 3;
  const float scale = 0.17677669529663687f;  // 1/sqrt(HD)

  const size_t base = (size_t)b * Qc * Dc + h * HDc;

  // ---- pass 1: scores -> S ----
  int qrow = q0 + n16; if (qrow > Qc - 1) qrow = Qc - 1;  // clamp padded rows
  const float2* qp = reinterpret_cast<const float2*>(q + base + (size_t)qrow * Dc + khalf);

  for (int nt = 0; nt < 19; ++nt) {
    int key = nt * 16 + n16;
    int keyc = key > Qc - 1 ? Qc - 1 : key;
    const float* kp = k + base + (size_t)keyc * Dc;
    v8f c = {};
#pragma unroll
    for (int s = 0; s < 8; ++s) {
      float2 av = qp[s * 2];
      v2f a; a.x = av.x; a.y = av.y;
      v2f bb; bb.x = kp[s * 4 + khalf]; bb.y = kp[s * 4 + khalf + 1];
      c = wmma_f32_4(a, bb, c);
    }
#pragma unroll
    for (int r = 0; r < 8; ++r) {
      float val = (key < Qc) ? c[r] * scale : -1e30f;
      S[r + mhi][nt * 16 + n16] = val;
    }
  }
  __syncthreads();

  // ---- softmax: lanes 0..15 each own one row ----
  if (lane < 16) {
    float mx = -1e30f;
    for (int j = 0; j < Qc; ++j) mx = fmaxf(mx, S[lane][j]);
    float sum = 0.0f;
    for (int j = 0; j < 304; ++j) {
      float e = __expf(S[lane][j] - mx);
      S[lane][j] = e;
      sum += e;
    }
    float inv = 1.0f / sum;
    for (int j = 0; j < 304; ++j) S[lane][j] *= inv;
  }
  __syncthreads();

  // ---- pass 2: ctx = P @ V (two 16-col tiles over HD=32) ----
  for (int nt = 0; nt < 2; ++nt) {
    v8f c = {};
#pragma unroll 4
    for (int s = 0; s < 76; ++s) {
      int kk = s * 4 + khalf;
      v2f a; a.x = S[n16][kk]; a.y = S[n16][kk + 1];
      int k0 = kk     > Qc - 1 ? Qc - 1 : kk;      // P==0 there; clamp for safety
      int k1 = kk + 1 > Qc - 1 ? Qc - 1 : kk + 1;
      v2f bb;
      bb.x = v[base + (size_t)k0 * Dc + nt * 16 + n16];
      bb.y = v[base + (size_t)k1 * Dc + nt * 16 + n16];
      c = wmma_f32_4(a, bb, c);
    }
#pragma unroll
    for (int r = 0; r < 8; ++r) {
      int row = q0 + r + mhi;
      if (row < Qc)
        ctx[base + (size_t)row * Dc + nt * 16 + n16] = c[r];
    }
  }
}

// ---------------------------------------------------------------------------
// Elementwise add
// ---------------------------------------------------------------------------
__global__ void add_kernel(const float* __restrict__ a, const float* __restrict__ b,
                           float* __restrict__ o, int n) {
  int i = blockIdx.x * blockDim.x + threadIdx.x;
  if (i < n) o[i] = a[i] + b[i];
}

// ---------------------------------------------------------------------------
// wave32 reduction helper
// ---------------------------------------------------------------------------
__device__ inline float wave_sum(float v) {
#pragma unroll
  for (int o = 16; o > 0; o >>= 1) v += __shfl_xor(v, o, 32);
  return v;
}

// y = LN(x1 + x2) * g + b     (rows of 256, one wave per row)
__global__ __launch_bounds__(256) void add_ln_kernel(
    const float* __restrict__ x1, const float* __restrict__ x2,
    const float* __restrict__ g, const float* __restrict__ bb,
    float* __restrict__ out, int rows) {
  int row = blockIdx.x * 8 + (threadIdx.x >> 5);
  if (row >= rows) return;
  int lane = threadIdx.x & 31;
  const float* a = x1 + (size_t)row * Dc;
  const float* c = x2 + (size_t)row * Dc;
  float vals[8], s = 0.0f;
#pragma unroll
  for (int j = 0; j < 8; ++j) { int d = lane + 32 * j; vals[j] = a[d] + c[d]; s += vals[j]; }
  float mean = wave_sum(s) * (1.0f / 256.0f);
  float vs = 0.0f;
#pragma unroll
  for (int j = 0; j < 8; ++j) { float dd = vals[j] - mean; vs += dd * dd; }
  float inv = rsqrtf(wave_sum(vs) * (1.0f / 256.0f) + 1e-5f);
  float* o = out + (size_t)row * Dc;
#pragma unroll
  for (int j = 0; j < 8; ++j) {
    int d = lane + 32 * j;
    o[d] = (vals[j] - mean) * inv * g[d] + bb[d];
  }
}

// y = LN(sig(g1)*res + sig(g2)*ca)
__global__ __launch_bounds__(256) void gate_ln_kernel(
    const float* __restrict__ gates, const float* __restrict__ res,
    const float* __restrict__ ca, const float* __restrict__ g,
    const float* __restrict__ bb, float* __restrict__ out, int rows) {
  int row = blockIdx.x * 8 + (threadIdx.x >> 5);
  if (row >= rows) return;
  int lane = threadIdx.x & 31;
  const float* gr = gates + (size_t)row * 512;
  const float* rr = res + (size_t)row * Dc;
  const float* cr = ca + (size_t)row * Dc;
  float vals[8], s = 0.0f;
#pragma unroll
  for (int j = 0; j < 8; ++j) {
    int d = lane + 32 * j;
    float g1 = 1.0f / (1.0f + __expf(-gr[d]));
    float g2 = 1.0f / (1.0f + __expf(-gr[256 + d]));
    vals[j] = g1 * rr[d] + g2 * cr[d];
    s += vals[j];
  }
  float mean = wave_sum(s) * (1.0f / 256.0f);
  float vs = 0.0f;
#pragma unroll
  for (int j = 0; j < 8; ++j) { float dd = vals[j] - mean; vs += dd * dd; }
  float inv = rsqrtf(wave_sum(vs) * (1.0f / 256.0f) + 1e-5f);
  float* o = out + (size_t)row * Dc;
#pragma unroll
  for (int j = 0; j < 8; ++j) {
    int d = lane + 32 * j;
    o[d] = (vals[j] - mean) * inv * g[d] + bb[d];
  }
}

// y = LN(clip(x1 + x2))
__global__ __launch_bounds__(256) void final_ln_kernel(
    const float* __restrict__ x1, const float* __restrict__ x2,
    const float* __restrict__ g, const float* __restrict__ bb,
    float* __restrict__ out, int rows) {
  int row = blockIdx.x * 8 + (threadIdx.x >> 5);
  if (row >= rows) return;
  int lane = threadIdx.x & 31;
  const float clampv = 3.4028234663852886e+38f - 1000.0f;
  const float* a = x1 + (size_t)row * Dc;
  const float* c = x2 + (size_t)row * Dc;
  float vals[8], s = 0.0f;
#pragma unroll
  for (int j = 0; j < 8; ++j) {
    int d = lane + 32 * j;
    float v = a[d] + c[d];
    v = fminf(fmaxf(v, -clampv), clampv);
    vals[j] = v; s += v;
  }
  float mean = wave_sum(s) * (1.0f / 256.0f);
  float vs = 0.0f;
#pragma unroll
  for (int j = 0; j < 8; ++j) { float dd = vals[j] - mean; vs += dd * dd; }
  float inv = rsqrtf(wave_sum(vs) * (1.0f / 256.0f) + 1e-5f);
  float* o = out + (size_t)row * Dc;
#pragma unroll
  for (int j = 0; j < 8; ++j) {
    int d = lane + 32 * j;
    o[d] = (vals[j] - mean) * inv * g[d] + bb[d];
  }
}

// ---------------------------------------------------------------------------
// Sampling locations + attention-weight softmax (per (b,q,h) thread)
// ---------------------------------------------------------------------------
__global__ void loc_aw_kernel(const float* __restrict__ off_raw,
                              const float* __restrict__ attn_raw,
                              const float* __restrict__ refp,
                              float* __restrict__ locs, float* __restrict__ aw) {
  int t = blockIdx.x * blockDim.x + threadIdx.x;
  if (t >= Bc * Qc * NHc) return;
  int h = t & 7;
  int bq = t >> 3;
  float r0 = refp[bq * 4 + 0], r1 = refp[bq * 4 + 1];
  float r2 = refp[bq * 4 + 2], r3 = refp[bq * 4 + 3];
  const float* orow = off_raw + (size_t)bq * 192 + h * 24;
  const float* arow = attn_raw + (size_t)bq * 96 + h * 12;
  float* lrow = locs + (size_t)bq * 192 + h * 24;
  float* wrow = aw + (size_t)bq * 96 + h * 12;

  float a[12], mx = -1e30f;
#pragma unroll
  for (int p = 0; p < 12; ++p) { a[p] = arow[p]; mx = fmaxf(mx, a[p]); }
  float s = 0.0f;
#pragma unroll
  for (int p = 0; p < 12; ++p) { a[p] = __expf(a[p] - mx); s += a[p]; }
  float inv = 1.0f / s;
#pragma unroll
  for (int p = 0; p < 12; ++p) wrow[p] = a[p] * inv;
  // nps = 1/4 for every point, OFFSET_SCALE = 0.5 -> factor 0.125
#pragma unroll
  for (int p = 0; p < 12; ++p) {
    lrow[2 * p + 0] = r0 + orow[2 * p + 0] * 0.125f * r2;
    lrow[2 * p + 1] = r1 + orow[2 * p + 1] * 0.125f * r3;
  }
}

// ---------------------------------------------------------------------------
// Deformable bilinear sampling: one thread per (b,q,h,d) output scalar
// ---------------------------------------------------------------------------
__device__ inline float dsamp(const float* __restrict__ base, int sp, int H, int W,
                              int y, int x) {
  if (x < 0 || x >= W || y < 0 || y >= H) return 0.0f;
  return base[(size_t)(sp + y * W + x) * Dc];
}

__global__ __launch_bounds__(256) void deform_kernel(
    const float* __restrict__ locs, const float* __restrict__ aw,
    const float* __restrict__ enc, float* __restrict__ ca) {
  int t = blockIdx.x * 256 + threadIdx.x;
  int d = t & 31;
  int h = (t >> 5) & 7;
  int bq = t >> 8;
  if (bq >= Bc * Qc) return;
  int b = bq / Qc;
  const float* lrow = locs + (size_t)bq * 192 + h * 24;
  const float* wrow = aw + (size_t)bq * 96 + h * 12;
  const float* base = enc + (size_t)b * LTOT * Dc + h * HDc + d;

  const int ssp[3] = {0, 6400, 8000};
  const int dim[3] = {80, 40, 20};
  float acc = 0.0f;
#pragma unroll
  for (int l = 0; l < 3; ++l) {
    int H = dim[l], W = dim[l], sp = ssp[l];
#pragma unroll
    for (int p = 0; p < 4; ++p) {
      int pi = l * 4 + p;
      float x = lrow[pi * 2 + 0] * W - 0.5f;
      float y = lrow[pi * 2 + 1] * H - 0.5f;
      float x0f = floorf(x), y0f = floorf(y);
      float lx = x - x0f, ly = y - y0f;
      int x0 = (int)x0f, y0 = (int)y0f;
      float v00 = dsamp(base, sp, H, W, y0, x0);
      float v01 = dsamp(base, sp, H, W, y0, x0 + 1);
      float v10 = dsamp(base, sp, H, W, y0 + 1, x0);
      float v11 = dsamp(base, sp, H, W, y0 + 1, x0 + 1);
      float sampled = (v00 * (1.0f - lx) + v01 * lx) * (1.0f - ly) +
                      (v10 * (1.0f - lx) + v11 * lx) * ly;
      acc += sampled * wrow[pi];
    }
  }
  ca[(size_t)bq * Dc + h * HDc + d] = acc;
}

// ---------------------------------------------------------------------------
// Host launcher
// ---------------------------------------------------------------------------
extern "C" void kernel_launch(void* const* d_in, const int* in_sizes, int n_in,
                              void* d_out, int out_size, void* d_ws, size_t ws_size,
                              hipStream_t stream) {
  const float* hidden = (const float*)d_in[0];
  const float* pos    = (const float*)d_in[1];
  const float* refp   = (const float*)d_in[2];
  const float* enc    = (const float*)d_in[3];
  const float* Wq = (const float*)d_in[4];  const float* bq = (const float*)d_in[5];
  const float* Wk = (const float*)d_in[6];  const float* bk = (const float*)d_in[7];
  const float* Wv = (const float*)d_in[8];  const float* bv = (const float*)d_in[9];
  const float* Wo = (const float*)d_in[10]; const float* bo = (const float*)d_in[11];
  const float* ln1_g = (const float*)d_in[12]; const float* ln1_b = (const float*)d_in[13];
  const float* W_off = (const float*)d_in[14]; const float* b_off = (const float*)d_in[15];
  const float* W_attn = (const float*)d_in[16]; const float* b_attn = (const float*)d_in[17];
  const float* W_gate = (const float*)d_in[18]; const float* b_gate = (const float*)d_in[19];
  const float* gln_g = (const float*)d_in[20]; const float* gln_b = (const float*)d_in[21];
  const float* W_fc1 = (const float*)d_in[22]; const float* b_fc1 = (const float*)d_in[23];
  const float* W_fc2 = (const float*)d_in[24]; const float* b_fc2 = (const float*)d_in[25];
  const float* ln2_g = (const float*)d_in[26]; const float* ln2_b = (const float*)d_in[27];
  float* out = (float*)d_out;

  const int M = Bc * Qc;               // 9600
  const size_t S1 = (size_t)M * Dc;    // 2,457,600 floats
  float* ws = (float*)d_ws;
  // Arena (~112 MB total). Lifetimes verified against launch order below.
  float* A0 = ws;                  // qk_in -> o -> locs -> (free)
  float* A1 = A0 + S1;             // q -> h (residual, live to end)
  float* A2 = A1 + S1;             // k -> qc -> h2
  float* A3 = A2 + S1;             // v -> off_raw -> ffn2
  float* A4 = A3 + S1;             // ctx -> attn_raw -> ca
  float* A5 = A4 + S1;             // aw (9600*96)
  float* A6 = A5 + (size_t)M * 96; // gates (9600*512)
  float* A7 = A6 + (size_t)M * 512;// ffn1 (9600*1024)

  auto gemm = [&](const float* A, int lda, const float* W, int ldw,
                  const float* bias, float* C, int ldc, int N, int K,
                  int relu, int accum) {
    dim3 g(N / 16, M / 128);
    gemm_wmma_kernel<<<g, 256, 0, stream>>>(A, lda, W, ldw, bias, C, ldc,
                                            M, N, K, relu, accum);
  };

  const int nElem = (int)S1;
  // qk_in = hidden + pos
  add_kernel<<<(nElem + 255) / 256, 256, 0, stream>>>(hidden, pos, A0, nElem);
  // q, k, v projections
  gemm(A0, Dc, Wq, Dc, bq, A1, Dc, Dc, Dc, 0, 0);
  gemm(A0, Dc, Wk, Dc, bk, A2, Dc, Dc, Dc, 0, 0);
  gemm(hidden, Dc, Wv, Dc, bv, A3, Dc, Dc, Dc, 0, 0);
  // self-attention -> ctx (A4)
  attn_kernel<<<19 * NHc * Bc, 32, 0, stream>>>(A1, A2, A3, A4);
  // o = ctx @ Wo + bo -> A0
  gemm(A4, Dc, Wo, Dc, bo, A0, Dc, Dc, Dc, 0, 0);
  // h = LN(hidden + o) -> A1  (residual)
  add_ln_kernel<<<M / 8, 256, 0, stream>>>(hidden, A0, ln1_g, ln1_b, A1, M);
  // qc = h + pos -> A2
  add_kernel<<<(nElem + 255) / 256, 256, 0, stream>>>(A1, pos, A2, nElem);
  // offsets / attn-weight projections
  gemm(A2, Dc, W_off, 192, b_off, A3, 192, 192, Dc, 0, 0);
  gemm(A2, Dc, W_attn, 96, b_attn, A4, 96, 96, Dc, 0, 0);
  // sampling locations (A0) + softmaxed weights (A5)
  loc_aw_kernel<<<(Bc * Qc * NHc + 255) / 256, 256, 0, stream>>>(A3, A4, refp, A0, A5);
  // deformable sampling -> ca (A4)
  deform_kernel<<<(int)(S1 / 256), 256, 0, stream>>>(A0, A5, enc, A4);
  // gates = h @ Wg[:256] + b  ;  gates += ca @ Wg[256:]
  gemm(A1, Dc, W_gate, 512, b_gate, A6, 512, 512, Dc, 0, 0);
  gemm(A4, Dc, W_gate + 256 * 512, 512, nullptr, A6, 512, 512, Dc, 0, 1);
  // h2 = LN(sig(g1)*h + sig(g2)*ca) -> A2
  gate_ln_kernel<<<M / 8, 256, 0, stream>>>(A6, A1, A4, gln_g, gln_b, A2, M);
  // FFN
  gemm(A2, Dc, W_fc1, 1024, b_fc1, A7, 1024, 1024, Dc, 1, 0);
  gemm(A7, 1024, W_fc2, Dc, b_fc2, A3, Dc, Dc, 1024, 0, 0);
  // out = LN(clip(h2 + ffn2))
  final_ln_kernel<<<M / 8, 256, 0, stream>>>(A2, A3, ln2_g, ln2_b, out, M);
}